// ACE_89240830476767
// MI455X (gfx1250) — compile-verified
//
#include <hip/hip_runtime.h>
#include <math.h>

typedef __attribute__((ext_vector_type(2))) float v2f;
typedef __attribute__((ext_vector_type(8))) float v8f;

#if defined(__HIP_DEVICE_COMPILE__) && !__has_builtin(__builtin_amdgcn_wmma_f32_16x16x4_f32)
#error "missing __builtin_amdgcn_wmma_f32_16x16x4_f32 on this toolchain"
#endif

#define N_  32
#define T_  256
#define K_  8000
#define L_  64
#define CH_ 16              // t-chunks; each chunk covers T_/CH_ = 16 timesteps
#define TC_ (T_ / CH_)      // 16 timesteps per chunk
#define EPS_ 1e-10f

// ---------------------------------------------------------------------------
// Phase 1: partial column sums via WMMA.
// Grid (CH_, N_), block = 128 (4 waves). Wave w owns labels [16w,16w+16) of row
// n = blockIdx.y, timesteps [16*blockIdx.x, +16). A(16x4)=ones, B(4x16)=gathered
// probs -> D[m][j] = chunk sum for label j (rows redundant). Since A is all
// ones, any bijection of the 4 K-values across B's (vgpr, lane-half) slots is
// correct. All lane/block-variant address math is folded into ONE base pointer
// so the 8 gathers become immediate-offset loads in a single clause.
// Lanes 0..15 write D[0][j] to ws[chunk][n*64 + 16w + j] (coalesced).
// ---------------------------------------------------------------------------
__global__ __launch_bounds__(128) void ACE_partial_colsum(
    const float* __restrict__ probs,   // [N, T, K]
    const int*   __restrict__ targets, // [N, L]
    float*       __restrict__ ws)      // [CH_, N_*L_]
{
#if defined(__HIP_DEVICE_COMPILE__)
    const int chunk = blockIdx.x;
    const int n     = blockIdx.y;
    const int wave  = threadIdx.x >> 5;  // 0..3 -> label group
    const int lane  = threadIdx.x & 31;
    const int j     = lane & 15;         // label within group
    const int h     = lane >> 4;         // lane half -> K sub-rows

    const int pair  = n * L_ + wave * 16 + j;        // (n, l) flat index
    const int label = targets[pair];
    // base points at probs[n, t0 + 2h, label]; every gather below is a
    // compile-time immediate offset (max 13*K_*4 = 416000 B < 2^23).
    const float* base = probs + (size_t)n * T_ * K_
                              + (size_t)(chunk * TC_ + 2 * h) * K_
                              + (size_t)label;

    v2f a; a.x = 1.0f; a.y = 1.0f;
    v8f c = {};

    #pragma unroll
    for (int i = 0; i < TC_; i += 4) {   // i is a compile-time constant
        v2f b;
        b.x = base[(size_t)(i + 0) * K_];
        b.y = base[(size_t)(i + 1) * K_];
        c = __builtin_amdgcn_wmma_f32_16x16x4_f32(
                false, a, false, b, (short)0, c, false, false);
    }

    if (lane < 16)
        ws[(size_t)chunk * (N_ * L_) + pair] = c[0];
#endif
}

// ---------------------------------------------------------------------------
// Phase 2: single block; deterministic (no atomics). For each (n,l) pair sum
// the CH_ chunk partials, take log(mean), reduce, scale, write scalar.
// ---------------------------------------------------------------------------
__global__ __launch_bounds__(256) void ACE_finalize(
    const float* __restrict__ ws, float* __restrict__ out)
{
    __shared__ float red[256];
    const int tid = threadIdx.x;
    float acc = 0.0f;
    for (int p = tid; p < N_ * L_; p += 256) {
        float S = 0.0f;
        #pragma unroll
        for (int cc = 0; cc < CH_; ++cc)
            S += ws[(size_t)cc * (N_ * L_) + p];
        acc += logf((S + (float)T_ * EPS_) * (1.0f / (float)T_));
    }
    red[tid] = acc;
    __syncthreads();
    for (int s = 128; s > 0; s >>= 1) {
        if (tid < s) red[tid] += red[tid + s];
        __syncthreads();
    }
    if (tid == 0)
        out[0] = red[0] * (-1.0f / ((float)N_ * (float)T_));
}

// ---------------------------------------------------------------------------
// Fallback (tiny ws): single-kernel version, atomic accumulation into out.
// ---------------------------------------------------------------------------
__global__ void ACE_zero_out(float* out) { out[0] = 0.0f; }

__global__ __launch_bounds__(128) void ACE_fused_loss(
    const float* __restrict__ probs, const int* __restrict__ targets,
    float* __restrict__ out)
{
#if defined(__HIP_DEVICE_COMPILE__)
    const int n    = blockIdx.x;
    const int wave = threadIdx.x >> 5;
    const int lane = threadIdx.x & 31;
    const int j    = lane & 15;
    const int h    = lane >> 4;

    const int label = targets[n * L_ + wave * 16 + j];
    const float* base = probs + (size_t)n * T_ * K_ + (size_t)(2 * h) * K_
                              + (size_t)label;

    v2f a; a.x = 1.0f; a.y = 1.0f;
    v8f c = {};
    #pragma unroll 4
    for (int t = 0; t < T_; t += 4) {
        v2f b;
        b.x = base[(size_t)(t + 0) * K_];
        b.y = base[(size_t)(t + 1) * K_];
        c = __builtin_amdgcn_wmma_f32_16x16x4_f32(
                false, a, false, b, (short)0, c, false, false);
    }
    float term = 0.0f;
    if (lane < 16)
        term = logf((c[0] + (float)T_ * EPS_) * (1.0f / (float)T_));
    for (int off = 8; off >= 1; off >>= 1)
        term += __shfl_down(term, off, 32);
    if (lane == 0)
        atomicAdd(out, term * (-1.0f / ((float)N_ * (float)T_)));
#endif
}

extern "C" void kernel_launch(void* const* d_in, const int* in_sizes, int n_in,
                              void* d_out, int out_size, void* d_ws, size_t ws_size,
                              hipStream_t stream) {
    (void)in_sizes; (void)n_in; (void)out_size;
    const float* probs   = (const float*)d_in[0];
    const int*   targets = (const int*)d_in[1];   // harness delivers ints as int32
    float*       out     = (float*)d_out;

    const size_t ws_needed = (size_t)CH_ * N_ * L_ * sizeof(float); // 128 KB
    if (ws_size >= ws_needed && d_ws != nullptr) {
        float* ws = (float*)d_ws;
        dim3 grid(CH_, N_);
        ACE_partial_colsum<<<grid, 128, 0, stream>>>(probs, targets, ws);
        ACE_finalize<<<1, 256, 0, stream>>>(ws, out);
    } else {
        ACE_zero_out<<<1, 1, 0, stream>>>(out);
        ACE_fused_loss<<<N_, 128, 0, stream>>>(probs, targets, out);
    }
}